// GroCo_89240830476381
// MI455X (gfx1250) — compile-verified
//
#include <hip/hip_runtime.h>
#include <hip/hip_bf16.h>
#include <math.h>

// Problem constants (match reference)
#define BB 8192
#define DD 128
#define TOPK 10
#define NSORT 11

typedef float v2f __attribute__((ext_vector_type(2)));
typedef float v8f __attribute__((ext_vector_type(8)));

// ---------------------------------------------------------------------------
// Kernel 1: row-normalize both inputs, compute positive similarities.
// grid = B blocks, block = 128 threads (one per feature).
// ---------------------------------------------------------------------------
__global__ __launch_bounds__(128) void GroCo_norm_kernel(
    const float* __restrict__ a1, const float* __restrict__ a2,
    float* __restrict__ n1, float* __restrict__ n2, float* __restrict__ pos)
{
    __shared__ float s[128];
    const int row = blockIdx.x;
    const int t   = threadIdx.x;

    const float v1 = a1[(size_t)row * DD + t];
    const float v2 = a2[(size_t)row * DD + t];

    auto block_reduce = [&](float val) -> float {
        s[t] = val;
        __syncthreads();
        #pragma unroll
        for (int st = 64; st > 0; st >>= 1) {
            if (t < st) s[t] += s[t + st];
            __syncthreads();
        }
        float r = s[0];
        __syncthreads();
        return r;
    };

    const float ss1 = block_reduce(v1 * v1);
    const float ss2 = block_reduce(v2 * v2);
    const float u1 = v1 / fmaxf(sqrtf(ss1), 1e-8f);
    const float u2 = v2 / fmaxf(sqrtf(ss2), 1e-8f);
    n1[(size_t)row * DD + t] = u1;
    n2[(size_t)row * DD + t] = u2;

    const float dp = block_reduce(u1 * u2);
    if (t == 0) pos[row] = dp;
}

// ---------------------------------------------------------------------------
// Kernel 2: fused  sim = nd @ nd.T  (FP32 WMMA)  +  per-row top-10 (diag
// masked).  grid = (B/16, 2 dirs), block = 128 threads = 4 waves.
// All 4 waves share the block's 16 rows; each wave processes column tiles
// j = wave, wave+4, ... and keeps private top-10 lists; merged at the end.
// ---------------------------------------------------------------------------
__global__ __launch_bounds__(128) void GroCo_sim_topk_kernel(
    const float* __restrict__ n1, const float* __restrict__ n2,
    float* __restrict__ top)
{
    const int dir  = blockIdx.y;
    const float* __restrict__ nd = dir ? n2 : n1;
    const int m0   = blockIdx.x * 16;
    const int w    = threadIdx.x >> 5;   // wave id 0..3
    const int lane = threadIdx.x & 31;
    const int half = lane >> 4;          // 0: K low pair / cols 0-7, 1: high
    const int l16  = lane & 15;
    const int koff = half * 2;           // ISA 32-bit A/B fragment layout

    __shared__ float tiles[4][16 * DD];           // 32 KB: per-wave col tiles
    __shared__ float mergebuf[16][8][TOPK];       // 5 KB : partial top lists
    float* tile = &tiles[w][0];

    // A strip (rows m0..m0+15, all K) resident in registers.
    // v_wmma_f32_16x16x4_f32 A layout: lanes 0-15 hold M=lane K={kb,kb+1},
    // lanes 16-31 hold M=lane-16 K={kb+2,kb+3}.
    v2f A[32];
    const float* arow = nd + (size_t)(m0 + l16) * DD;
    #pragma unroll
    for (int i = 0; i < 32; ++i) {
        float2 t2 = *(const float2*)(arow + 4 * i + koff);
        A[i].x = t2.x;
        A[i].y = t2.y;
    }

    float tk[TOPK];                       // ascending: tk[0] = current min
    #pragma unroll
    for (int i = 0; i < TOPK; ++i) tk[i] = -INFINITY;

    const int grow = m0 + l16;            // global row this lane scans for

    for (int j = w; j < BB / 16; j += 4) {  // 128 uniform iterations per wave
        const int c0 = j * 16;

        // ---- stage 16x128 column strip into this wave's LDS tile ----
        #pragma unroll
        for (int r = 0; r < 16; ++r) {
            float4 v = *(const float4*)(nd + (size_t)(c0 + r) * DD + lane * 4);
            *(float4*)(tile + r * DD + lane * 4) = v;
        }
        __syncthreads();   // uniform: all waves have identical trip counts

        // ---- 16x16x128 via 32 x v_wmma_f32_16x16x4_f32 ----
        v8f c = {0.f, 0.f, 0.f, 0.f, 0.f, 0.f, 0.f, 0.f};
        #pragma unroll
        for (int i = 0; i < 32; ++i) {
            // B layout (K x N = 4 x 16): lanes 0-15: {K=kb,kb+1} of col N=lane,
            // lanes 16-31: {K=kb+2,kb+3} of col N=lane-16.
            float2 bt = *(const float2*)(tile + l16 * DD + 4 * i + koff);
            v2f b; b.x = bt.x; b.y = bt.y;
            c = __builtin_amdgcn_wmma_f32_16x16x4_f32(
                    false, A[i], false, b, (short)0, c, false, false);
        }
        __syncthreads();

        // ---- reshard C by row through LDS (C layout: VGPR v holds row v
        //      (lanes 0-15) / row v+8 (lanes 16-31), col = lane%16) ----
        #pragma unroll
        for (int v = 0; v < 8; ++v) {
            tile[(v + half * 8) * 16 + l16] = c[v];
        }
        __syncthreads();

        // ---- top-k update: lane owns row l16, cols half*8 .. half*8+7 ----
        const float* crow = tile + l16 * 16 + half * 8;
        #pragma unroll
        for (int cc = 0; cc < 8; ++cc) {
            float v = crow[cc];
            const int col = c0 + half * 8 + cc;
            if (col == grow) v = -INFINITY;       // mask diagonal
            if (v > tk[0]) {                      // insert, keep ascending
                tk[0] = v;
                #pragma unroll
                for (int k = 0; k < TOPK - 1; ++k) {
                    if (tk[k] > tk[k + 1]) {
                        float tt = tk[k]; tk[k] = tk[k + 1]; tk[k + 1] = tt;
                    }
                }
            }
        }
        __syncthreads();
    }

    // ---- merge 8 partial lists per row ----
    #pragma unroll
    for (int i = 0; i < TOPK; ++i) mergebuf[l16][w * 2 + half][i] = tk[i];
    __syncthreads();

    if (threadIdx.x < 16) {
        const int ro = threadIdx.x;
        float f[TOPK];
        #pragma unroll
        for (int i = 0; i < TOPK; ++i) f[i] = -INFINITY;
        for (int lst = 0; lst < 8; ++lst) {
            #pragma unroll
            for (int i = 0; i < TOPK; ++i) {
                float v = mergebuf[ro][lst][i];
                if (v > f[0]) {
                    f[0] = v;
                    #pragma unroll
                    for (int k = 0; k < TOPK - 1; ++k) {
                        if (f[k] > f[k + 1]) {
                            float tt = f[k]; f[k] = f[k + 1]; f[k + 1] = tt;
                        }
                    }
                }
            }
        }
        float* dst = top + ((size_t)dir * BB + (m0 + ro)) * TOPK;
        #pragma unroll
        for (int i = 0; i < TOPK; ++i) dst[i] = f[i];  // ascending order
    }
}

// ---------------------------------------------------------------------------
// Kernel 3: per-(row,dir) differentiable odd-even sort (n=11) + BCE partials.
// One thread per (row, dir); P matrix fully unrolled into registers.
// ---------------------------------------------------------------------------
__global__ __launch_bounds__(256) void GroCo_sortloss_kernel(
    const float* __restrict__ top, const float* __restrict__ pos,
    float* __restrict__ rowloss)
{
    const int idx = blockIdx.x * 256 + threadIdx.x;   // 0 .. 2*B-1
    const int row = idx & (BB - 1);

    const float INVPI = 0.3183098861837907f;

    float x[NSORT];
    #pragma unroll
    for (int i = 0; i < TOPK; ++i) x[i] = top[(size_t)idx * TOPK + i];
    x[TOPK] = pos[row];

    float P[NSORT][NSORT];
    #pragma unroll
    for (int i = 0; i < NSORT; ++i)
        #pragma unroll
        for (int j = 0; j < NSORT; ++j)
            P[i][j] = (i == j) ? 1.0f : 0.0f;

    #pragma unroll
    for (int layer = 0; layer < NSORT; ++layer) {
        const int off = layer & 1;
        #pragma unroll
        for (int a = off; a + 1 < NSORT; a += 2) {
            const float xa = x[a], xb = x[a + 1];
            const float alpha = atanf(1.0f * (xb - xa)) * INVPI + 0.5f;
            const float beta  = 1.0f - alpha;
            x[a]     = alpha * xa + beta * xb;
            x[a + 1] = beta * xa + alpha * xb;
            #pragma unroll
            for (int r = 0; r < NSORT; ++r) {
                const float pa = P[r][a], pb = P[r][a + 1];
                P[r][a]     = alpha * pa + beta * pb;
                P[r][a + 1] = beta * pa + alpha * pb;
            }
        }
    }

    float acc = 0.0f;
    #pragma unroll
    for (int i = 0; i < NSORT; ++i) {
        float pneg = 0.0f;
        #pragma unroll
        for (int j = 0; j < TOPK; ++j) pneg += P[i][j];
        pneg = fminf(fmaxf(pneg, 0.0f), 1.0f);
        float ppos = fminf(fmaxf(P[i][TOPK], 0.0f), 1.0f);

        const float tneg = (i < TOPK) ? 1.0f : 0.0f;
        const float tpos = 1.0f - tneg;

        const float lpp  = fmaxf(logf(ppos), -100.0f);
        const float l1pp = fmaxf(logf(1.0f - ppos), -100.0f);
        const float lpn  = fmaxf(logf(pneg), -100.0f);
        const float l1pn = fmaxf(logf(1.0f - pneg), -100.0f);

        acc += -(tpos * lpp + (1.0f - tpos) * l1pp);   // bce(p_pos, t_pos)
        acc += -(tneg * lpn + (1.0f - tneg) * l1pn);   // bce(p_neg, t_neg)
    }
    rowloss[idx] = acc;
}

// ---------------------------------------------------------------------------
// Kernel 4: deterministic reduction of 2*B partials -> scalar loss.
// final = 0.5*(bce1p+bce1n)/2 + 0.5*(bce2p+bce2n)/2 = sum * 0.25/(B*11)
// ---------------------------------------------------------------------------
__global__ __launch_bounds__(256) void GroCo_reduce_kernel(
    const float* __restrict__ rowloss, float* __restrict__ out)
{
    __shared__ float s[256];
    float acc = 0.0f;
    for (int i = threadIdx.x; i < 2 * BB; i += 256) acc += rowloss[i];
    s[threadIdx.x] = acc;
    __syncthreads();
    #pragma unroll
    for (int st = 128; st > 0; st >>= 1) {
        if (threadIdx.x < st) s[threadIdx.x] += s[threadIdx.x + st];
        __syncthreads();
    }
    if (threadIdx.x == 0)
        out[0] = s[0] * (0.25f / ((float)BB * (float)NSORT));
}

// ---------------------------------------------------------------------------
extern "C" void kernel_launch(void* const* d_in, const int* in_sizes, int n_in,
                              void* d_out, int out_size, void* d_ws, size_t ws_size,
                              hipStream_t stream)
{
    const float* a1 = (const float*)d_in[0];
    const float* a2 = (const float*)d_in[1];

    float* ws      = (float*)d_ws;
    float* n1      = ws;                          // B*D
    float* n2      = n1 + (size_t)BB * DD;        // B*D
    float* pos     = n2 + (size_t)BB * DD;        // B
    float* top     = pos + BB;                    // 2*B*TOPK
    float* rowloss = top + (size_t)2 * BB * TOPK; // 2*B

    GroCo_norm_kernel<<<BB, 128, 0, stream>>>(a1, a2, n1, n2, pos);

    dim3 g(BB / 16, 2);
    GroCo_sim_topk_kernel<<<g, 128, 0, stream>>>(n1, n2, top);

    GroCo_sortloss_kernel<<<(2 * BB) / 256, 256, 0, stream>>>(top, pos, rowloss);

    GroCo_reduce_kernel<<<1, 256, 0, stream>>>(rowloss, (float*)d_out);
}